// GDR_81819126989199
// MI455X (gfx1250) — compile-verified
//
#include <hip/hip_runtime.h>
#include <math.h>

#define B_ 8
#define L_ 2048
#define D_ 128
#define M_ (B_*L_)        // 16384 rows
#define NPROJ 896         // 512 qkv + 256 gate-eff + 128 residual
#define VPK 640           // packed per-step vectors: q,k,alpha,beta*k,v

typedef __attribute__((ext_vector_type(2))) float v2f;
typedef __attribute__((ext_vector_type(8))) float v8f;

__device__ __forceinline__ float sigmoidf_(float x){ return 1.0f/(1.0f + __expf(-x)); }

// low 32 bits of a generic pointer to LDS == LDS byte offset (ISA §10.2 aperture map)
__device__ __forceinline__ unsigned lds_off_(const void* p) {
  return (unsigned)(unsigned long long)(uintptr_t)p;
}

// ---------------------------------------------------------------------------
// Kernel 0: W_geff[256,128] = W_gate[256,128] @ W_in[384:512,128]   (f32 WMMA)
// ---------------------------------------------------------------------------
__global__ void kgeff(const float* __restrict__ Wg, const float* __restrict__ WinG,
                      float* __restrict__ Wge) {
  const int lane = threadIdx.x & 31, wave = threadIdx.x >> 5;
  const int lm = lane & 15, lh = lane >> 4;
  const int row0 = blockIdx.x * 16;   // 16 blocks cover 256 rows
  const int col0 = wave * 16;         // 8 waves cover 128 cols
  v8f acc = {};
  const float* pa = Wg   + (row0 + lm) * D_;   // A[m][k], row-major
  const float* pb = WinG + (col0 + lm);        // B[k][n], row-major in k
  #pragma unroll
  for (int k0 = 0; k0 < D_; k0 += 4) {
    const int kk = k0 + 2*lh;
    v2f a = *(const v2f*)(pa + kk);
    v2f b; b.x = pb[(size_t)kk*D_]; b.y = pb[(size_t)(kk+1)*D_];
    acc = __builtin_amdgcn_wmma_f32_16x16x4_f32(false, a, false, b, (short)0, acc, false, false);
  }
  #pragma unroll
  for (int r = 0; r < 8; ++r)
    Wge[(row0 + r + 8*lh)*D_ + col0 + lm] = acc[r];
}

// ---------------------------------------------------------------------------
// Kernel 1: proj[M,896] = x[M,128] @ [W_in ; W_geff ; W_res]^T    (f32 WMMA)
// ---------------------------------------------------------------------------
__global__ void kproj(const float* __restrict__ x, const float* __restrict__ Win,
                      const float* __restrict__ Wge, const float* __restrict__ Wres,
                      float* __restrict__ proj) {
  const int lane = threadIdx.x & 31, wave = threadIdx.x >> 5;
  const int lm = lane & 15, lh = lane >> 4;
  const int row0 = blockIdx.x * 16;                 // M/16 = 1024
  const int col0 = (blockIdx.y * 8 + wave) * 16;    // 7*8 waves -> 896 cols
  const float* Wp; int nb;
  if (col0 < 512)      { Wp = Win;  nb = col0;       }
  else if (col0 < 768) { Wp = Wge;  nb = col0 - 512; }
  else                 { Wp = Wres; nb = col0 - 768; }
  v8f acc = {};
  const float* pa = x  + (size_t)(row0 + lm) * D_;
  const float* pb = Wp + (size_t)(nb  + lm) * D_;   // W row-major [n][k] -> contiguous k
  #pragma unroll
  for (int k0 = 0; k0 < D_; k0 += 4) {
    const int kk = k0 + 2*lh;
    v2f a = *(const v2f*)(pa + kk);
    v2f b = *(const v2f*)(pb + kk);
    acc = __builtin_amdgcn_wmma_f32_16x16x4_f32(false, a, false, b, (short)0, acc, false, false);
  }
  #pragma unroll
  for (int r = 0; r < 8; ++r)
    proj[(size_t)(row0 + r + 8*lh)*NPROJ + col0 + lm] = acc[r];
}

// ---------------------------------------------------------------------------
// Kernel 2: conv3 + sigmoid + L2-norm for q,k; sigmoid gates; pack per-step
// vectors {q, k, alpha, beta*k, v} contiguously for the scan's async prefetch
// ---------------------------------------------------------------------------
__global__ void kprep(const float* __restrict__ proj,
                      const float* __restrict__ qw, const float* __restrict__ qb,
                      const float* __restrict__ kw, const float* __restrict__ kb,
                      float* __restrict__ pk) {
  __shared__ float red[D_];
  const int idx = blockIdx.x;           // b*L + l
  const int l   = idx & (L_ - 1);
  const int d   = threadIdx.x;
  const float* rowp = proj + (size_t)idx * NPROJ;
  float qc = qb[d], kc = kb[d];
  #pragma unroll
  for (int t = -1; t <= 1; ++t) {       // cross-correlation, zero pad in L
    const int ll = l + t;
    if (ll >= 0 && ll < L_) {
      const float* rp = rowp + (long)t * NPROJ;
      qc += qw[d*3 + (t+1)] * rp[d];
      kc += kw[d*3 + (t+1)] * rp[D_ + d];
    }
  }
  qc = sigmoidf_(qc); kc = sigmoidf_(kc);
  red[d] = qc*qc; __syncthreads();
  for (int s = 64; s > 0; s >>= 1) { if (d < s) red[d] += red[d+s]; __syncthreads(); }
  const float nq = fmaxf(sqrtf(red[0]), 1e-12f);
  __syncthreads();
  red[d] = kc*kc; __syncthreads();
  for (int s = 64; s > 0; s >>= 1) { if (d < s) red[d] += red[d+s]; __syncthreads(); }
  const float nk = fmaxf(sqrtf(red[0]), 1e-12f);
  const float qn = qc / nq, kn = kc / nk;
  const size_t po = (size_t)idx * VPK;
  pk[po +       d] = qn;
  pk[po + 128 + d] = kn;
  pk[po + 256 + d] = sigmoidf_(rowp[512 + d]);           // alpha
  pk[po + 384 + d] = sigmoidf_(rowp[640 + d]) * kn;      // beta * k
  pk[po + 512 + d] = rowp[256 + d];                      // v
}

// ---------------------------------------------------------------------------
// Kernel 3: sequential delta-rule scan, one workgroup per batch, S in LDS,
// next step's operand vectors prefetched with GLOBAL_LOAD_ASYNC_TO_LDS_B128
// into a double buffer (latency of the 2.5KB/step fetch hidden behind compute)
// ---------------------------------------------------------------------------
__global__ void kscan(const float* __restrict__ state, const float* __restrict__ pk,
                      float* __restrict__ ocore, float* __restrict__ sfin) {
  extern __shared__ float smem[];
  float* S    = smem;                 // 128 x 129 (padded pitch -> conflict-free cols)
  float* vbuf = S + 128*129;          // 2 x 640 double-buffered step vectors
  float* Sk   = vbuf + 2*VPK;         // 128
  float* ps   = Sk + D_;              // 512 partials
  const int b = blockIdx.x;
  const int t = threadIdx.x;          // 512 threads = 16 waves
  // ---- load initial state
  for (int e = 0; e < 32; ++e) {
    const int i2 = e*512 + t;
    S[(i2>>7)*129 + (i2 & 127)] = state[(size_t)b*D_*D_ + i2];
  }
  // ---- prefetch step 0 into buffer 0 (160 lanes x 16B = 640 floats)
  if (t < 160) {
    const unsigned lds = lds_off_(vbuf) + t*16u;
    const unsigned long long ga =
        (unsigned long long)(uintptr_t)(pk + (size_t)b*L_*VPK) + (unsigned long long)t*16u;
    asm volatile("global_load_async_to_lds_b128 %0, %1, off" :: "v"(lds), "v"(ga) : "memory");
  }
  asm volatile("s_wait_asynccnt 0" ::: "memory");
  __syncthreads();

  const int row = t >> 2, j0 = (t & 3) * 32;
  int cur = 0;
  for (int l = 0; l < L_; ++l) {
    const size_t base = (size_t)b*L_ + l;
    const float* vc  = vbuf + cur*VPK;
    const float* qv  = vc;
    const float* kv  = vc + 128;
    const float* av  = vc + 256;
    const float* bkv = vc + 384;
    const float* vv  = vc + 512;
    // ---- kick off async prefetch of step l+1 into the other buffer
    if (l + 1 < L_ && t < 160) {
      const unsigned lds = lds_off_(vbuf) + (cur^1)*(VPK*4u) + t*16u;
      const unsigned long long ga =
          (unsigned long long)(uintptr_t)(pk + (base+1)*VPK) + (unsigned long long)t*16u;
      asm volatile("global_load_async_to_lds_b128 %0, %1, off" :: "v"(lds), "v"(ga) : "memory");
    }
    // ---- Sk[i] = sum_j S[i][j] * k[j]
    {
      float s = 0.f;
      const float* srow = S + row*129 + j0;
      #pragma unroll
      for (int j = 0; j < 32; ++j) s += srow[j] * kv[j0 + j];
      ps[t] = s;
    }
    __syncthreads();
    if (t < D_) Sk[t] = ps[t*4] + ps[t*4+1] + ps[t*4+2] + ps[t*4+3];
    __syncthreads();
    // ---- S[i][j] = a[j]*(S[i][j] - Sk[i]*bk[j]) + v[i]*bk[j]
    {
      const float ski = Sk[row], vi = vv[row];
      float* srow = S + row*129 + j0;
      #pragma unroll
      for (int j = 0; j < 32; ++j) {
        const float bkj = bkv[j0 + j];
        srow[j] = av[j0 + j]*(srow[j] - ski*bkj) + vi*bkj;
      }
    }
    __syncthreads();
    // ---- out[j] = silu(sum_i q[i]*S[i][j])  (uses updated S)
    {
      float s = 0.f;
      #pragma unroll
      for (int i = 0; i < 32; ++i) s += qv[j0 + i] * S[(j0 + i)*129 + row];
      ps[t] = s;
    }
    __syncthreads();
    if (t < D_) {
      const float o = ps[t*4] + ps[t*4+1] + ps[t*4+2] + ps[t*4+3];
      ocore[base*D_ + t] = o * sigmoidf_(o);
    }
    // ---- drain prefetch, publish buffer to whole workgroup, swap
    asm volatile("s_wait_asynccnt 0" ::: "memory");
    __syncthreads();
    cur ^= 1;
  }
  for (int e = 0; e < 32; ++e) {
    const int i2 = e*512 + t;
    sfin[(size_t)b*D_*D_ + i2] = S[(i2>>7)*129 + (i2 & 127)];
  }
}

// ---------------------------------------------------------------------------
// Kernel 4: out = RMSnorm(ocore @ W_out.T)*rms_w + residual   (f32 WMMA)
// one block = 16 rows x all 128 cols -> RMS fused in-block
// ---------------------------------------------------------------------------
__global__ void kout(const float* __restrict__ ocore, const float* __restrict__ Wout,
                     const float* __restrict__ rmsw, const float* __restrict__ proj,
                     float* __restrict__ out) {
  __shared__ float buf[16*136];
  __shared__ float ps[16*16];
  __shared__ float scale[16];
  const int lane = threadIdx.x & 31, wave = threadIdx.x >> 5;
  const int lm = lane & 15, lh = lane >> 4;
  const int row0 = blockIdx.x * 16;
  const int col0 = wave * 16;
  v8f acc = {};
  const float* pa = ocore + (size_t)(row0 + lm) * D_;
  const float* pb = Wout  + (size_t)(col0 + lm) * D_;
  #pragma unroll
  for (int k0 = 0; k0 < D_; k0 += 4) {
    const int kk = k0 + 2*lh;
    v2f a = *(const v2f*)(pa + kk);
    v2f b = *(const v2f*)(pb + kk);
    acc = __builtin_amdgcn_wmma_f32_16x16x4_f32(false, a, false, b, (short)0, acc, false, false);
  }
  #pragma unroll
  for (int r = 0; r < 8; ++r) buf[(r + 8*lh)*136 + col0 + lm] = acc[r];
  __syncthreads();
  const int t = threadIdx.x;           // 256
  {
    const int r = t >> 4, seg = t & 15;
    float s = 0.f;
    #pragma unroll
    for (int c = 0; c < 8; ++c) { const float v = buf[r*136 + seg*8 + c]; s += v*v; }
    ps[r*16 + seg] = s;
  }
  __syncthreads();
  if (t < 16) {
    float s = 0.f;
    #pragma unroll
    for (int c = 0; c < 16; ++c) s += ps[t*16 + c];
    scale[t] = rsqrtf(s * (1.0f/128.0f) + 1e-6f);
  }
  __syncthreads();
  #pragma unroll
  for (int e = 0; e < 8; ++e) {
    const int idx = e*256 + t;
    const int r = idx >> 7, c = idx & 127;
    const size_t m = (size_t)(row0 + r);
    out[m*D_ + c] = buf[r*136 + c]*scale[r]*rmsw[c] + proj[m*NPROJ + 768 + c];
  }
}

// ---------------------------------------------------------------------------
extern "C" void kernel_launch(void* const* d_in, const int* in_sizes, int n_in,
                              void* d_out, int out_size, void* d_ws, size_t ws_size,
                              hipStream_t stream) {
  const float* x     = (const float*)d_in[0];
  const float* state = (const float*)d_in[1];
  const float* Win   = (const float*)d_in[2];
  const float* Wg    = (const float*)d_in[3];
  const float* Wout  = (const float*)d_in[4];
  const float* Wres  = (const float*)d_in[5];
  const float* qcw   = (const float*)d_in[6];
  const float* qcb   = (const float*)d_in[7];
  const float* kcw   = (const float*)d_in[8];
  const float* kcb   = (const float*)d_in[9];
  const float* rmsw  = (const float*)d_in[10];

  float* out  = (float*)d_out;               // [M, D]
  float* sfin = out + (size_t)M_ * D_;       // [B, D, D]

  float* ws   = (float*)d_ws;                // ~109 MB total
  size_t o = 0;
  float* proj = ws + o;  o += (size_t)M_ * NPROJ;
  float* Wge  = ws + o;  o += 256 * 128;
  float* pk   = ws + o;  o += (size_t)M_ * VPK;
  float* oc   = ws + o;  o += (size_t)M_ * D_;

  kgeff<<<dim3(16),        256, 0, stream>>>(Wg, Win + 384*128, Wge);
  kproj<<<dim3(M_/16, 7),  256, 0, stream>>>(x, Win, Wge, Wres, proj);
  kprep<<<dim3(M_),        128, 0, stream>>>(proj, qcw, qcb, kcw, kcb, pk);
  const size_t smem = (size_t)(128*129 + 2*VPK + D_ + 512) * sizeof(float);  // ~72 KB
  kscan<<<dim3(B_),        512, smem, stream>>>(state, pk, oc, sfin);
  kout<<<dim3(M_/16),      256, 0, stream>>>(oc, Wout, rmsw, proj, out);
}